// ProsodyExtractor_14190571946368
// MI455X (gfx1250) — compile-verified
//
#include <hip/hip_runtime.h>
#include <hip/hip_bf16.h>

// ---------------------------------------------------------------------------
// ProsodyExtractor for MI455X (gfx1250), fp32 WMMA (V_WMMA_F32_16X16X4_F32).
// B=32, P=192, DMAX=48, DMEL=80, H=64, 3H=192, T=P*DMAX=9216, BP=6144.
// One wave32 = one 16-sequence tile. 4 waves/block, 96 blocks = 384 tiles.
// ---------------------------------------------------------------------------

#define BQ    32
#define PQ    192
#define DMAXQ 48
#define DMELQ 80
#define HQ    64
#define TQ    (PQ * DMAXQ)   // 9216
#define G3    192            // 3*H
#define WAVES 4
#define BN_EPS 1e-5f

typedef __attribute__((ext_vector_type(2))) float v2f;
typedef __attribute__((ext_vector_type(8))) float v8f;

// D = A(16x4) * B(4x16) + C(16x16), all fp32, wave32.
// arg pattern mirrors probe-confirmed 8-arg form of 16x16x{4,32} builtins:
// (neg_a, A, neg_b, B, c_mod, C, reuse_a, reuse_b)
__device__ __forceinline__ v8f wmma4(v2f a, v2f b, v8f c) {
  return __builtin_amdgcn_wmma_f32_16x16x4_f32(false, a, false, b, (short)0, c,
                                               false, false);
}

__device__ __forceinline__ float sigmoidf_fast(float x) {
  return 1.0f / (1.0f + __expf(-x));
}
__device__ __forceinline__ float tanhf_fast(float x) {
  // 2*sigmoid(2x)-1 : saturates correctly at +-1, no inf/inf NaN.
  return 2.0f / (1.0f + __expf(-2.0f * x)) - 1.0f;
}

// LDS float offsets
#define LDS_WIH   0                    // 192*80 = 15360  (Wih_f, later Wih_b)
#define LDS_WHH   15360                // 192*64 = 12288  (Whh_f)
#define LDS_BIAS  27648                // 4*192  =   768  [bihf|bhhf|bihb|bhhb]
#define LDS_HBUF  28416                // WAVES * 16*64 = 4096
#define LDS_FLOATS 32512               // 130,048 bytes

__global__ __launch_bounds__(WAVES * 32, 1)
void prosody_gru_kernel(const float* __restrict__ mels,      // (B,T,80)
                        const int*   __restrict__ durations, // (B,P)
                        const float* __restrict__ conv_w,    // (2)
                        const float* __restrict__ conv_b,    // (2)
                        const float* __restrict__ bn_g,      // (2)
                        const float* __restrict__ bn_b,      // (2)
                        const float* __restrict__ bn_m,      // (2)
                        const float* __restrict__ bn_v,      // (2)
                        const float* __restrict__ W_ih_f,    // (192,80)
                        const float* __restrict__ W_hh_f,    // (192,64)
                        const float* __restrict__ b_ih_f,    // (192)
                        const float* __restrict__ b_hh_f,    // (192)
                        const float* __restrict__ W_ih_b,    // (192,80)
                        const float* __restrict__ b_ih_b,    // (192)
                        const float* __restrict__ b_hh_b,    // (192)
                        float* __restrict__ out)             // (B,P,128)
{
  extern __shared__ float lds[];
  float* sWih  = lds + LDS_WIH;
  float* sWhh  = lds + LDS_WHH;
  float* sBias = lds + LDS_BIAS;

  const int tid = threadIdx.x;
  const int nthreads = WAVES * 32;

  // ---- cooperative LDS stage: forward weights + all biases ----------------
  for (int i = tid; i < G3 * DMELQ; i += nthreads) sWih[i] = W_ih_f[i];
  for (int i = tid; i < G3 * HQ;    i += nthreads) sWhh[i] = W_hh_f[i];
  for (int i = tid; i < G3; i += nthreads) {
    sBias[i]           = b_ih_f[i];
    sBias[G3 + i]      = b_hh_f[i];
    sBias[2 * G3 + i]  = b_ih_b[i];
    sBias[3 * G3 + i]  = b_hh_b[i];
  }
  __syncthreads();

  const int wave = tid >> 5;
  const int lane = tid & 31;
  const int m    = lane & 15;   // A-layout row / C-layout column within tile
  const int hi   = lane >> 4;   // half-wave select
  const int koff = hi * 2;      // K sub-offset for A/B operands

  const int tile = blockIdx.x * WAVES + wave;
  const int seq0 = tile * 16;
  const int b    = seq0 / PQ;   // 192 % 16 == 0 -> tile never crosses batch
  const int p0   = seq0 % PQ;

  // ---- fused conv + BN affine coefficients (two relu(a*x+c) layers) -------
  const float g0 = bn_g[0] * rsqrtf(bn_v[0] + BN_EPS);
  const float g1 = bn_g[1] * rsqrtf(bn_v[1] + BN_EPS);
  const float a0 = conv_w[0] * g0, c0 = (conv_b[0] - bn_m[0]) * g0 + bn_b[0];
  const float a1 = conv_w[1] * g1, c1 = (conv_b[1] - bn_m[1]) * g1 + bn_b[1];

  // ---- segment starts via in-wave cumsum of durations ---------------------
  const int* durB = durations + b * PQ;
  int base = 0;
  for (int q0 = 0; q0 < p0; q0 += 32) {
    int v = (q0 + lane < p0) ? durB[q0 + lane] : 0;
    #pragma unroll
    for (int off = 16; off > 0; off >>= 1) v += __shfl_xor(v, off, 32);
    base += v;                                    // all lanes hold the sum
  }
  const int dm = durB[p0 + m];                    // dur for this A-layout row
  int inc = dm;                                   // inclusive scan, width 16
  #pragma unroll
  for (int off = 1; off < 16; off <<= 1) {
    int t2 = __shfl_up(inc, off, 16);
    if (m >= off) inc += t2;
  }
  const int start_m = base + inc - dm;            // exclusive prefix

  int durD[8];                                    // durs in C/D row layout
  #pragma unroll
  for (int v = 0; v < 8; v++) durD[v] = durB[p0 + v + 8 * hi];

  const float* melsB = mels + (size_t)b * TQ * DMELQ;

  // ---- bias registers (C-init values; column n = 16*tile + m) -------------
  float biasRZ[8], biasIN[4], biasHN[4];
  #pragma unroll
  for (int tt = 0; tt < 8; tt++) {
    int n = tt * 16 + m;
    biasRZ[tt] = sBias[n] + sBias[G3 + n];        // b_ih + b_hh fused for r,z
  }
  #pragma unroll
  for (int j = 0; j < 4; j++) {
    int n = 128 + j * 16 + m;
    biasIN[j] = sBias[n];                         // b_ih (n-gate, i-path)
    biasHN[j] = sBias[G3 + n];                    // b_hh (n-gate, h-path)
  }

  // ---- per-wave h re-layout buffer (16x64) --------------------------------
  float* hb = lds + LDS_HBUF + wave * (16 * HQ);
  for (int i = lane; i < 16 * HQ; i += 32) hb[i] = 0.0f;

  float hreg[4][8];                               // h in C/D layout
  #pragma unroll
  for (int j = 0; j < 4; j++)
    #pragma unroll
    for (int v = 0; v < 8; v++) hreg[j][v] = 0.0f;

  v8f accRZ[8], accIN[4], accHN[4];

  // ======================= forward GRU over 48 steps =======================
  #pragma unroll 1
  for (int t = 0; t < DMAXQ; t++) {
    #pragma unroll
    for (int tt = 0; tt < 8; tt++)
      #pragma unroll
      for (int v = 0; v < 8; v++) accRZ[tt][v] = biasRZ[tt];
    #pragma unroll
    for (int j = 0; j < 4; j++)
      #pragma unroll
      for (int v = 0; v < 8; v++) { accIN[j][v] = biasIN[j]; accHN[j][v] = biasHN[j]; }

    // ---- gi = x_t @ Wih^T : A streamed from global with fused affine ----
    int frame = start_m + t; frame = frame < (TQ - 1) ? frame : (TQ - 1);
    const float* xrow = melsB + (size_t)frame * DMELQ;
    #pragma unroll
    for (int k0 = 0; k0 < DMELQ; k0 += 4) {
      v2f xa = *(const v2f*)(xrow + k0 + koff);
      xa.x = fmaxf(0.0f, fmaf(a1, fmaxf(0.0f, fmaf(a0, xa.x, c0)), c1));
      xa.y = fmaxf(0.0f, fmaf(a1, fmaxf(0.0f, fmaf(a0, xa.y, c0)), c1));
      #pragma unroll
      for (int tt = 0; tt < 8; tt++) {
        v2f wb = *(const v2f*)(sWih + (tt * 16 + m) * DMELQ + k0 + koff);
        accRZ[tt] = wmma4(xa, wb, accRZ[tt]);
      }
      #pragma unroll
      for (int j = 0; j < 4; j++) {
        v2f wb = *(const v2f*)(sWih + ((8 + j) * 16 + m) * DMELQ + k0 + koff);
        accIN[j] = wmma4(xa, wb, accIN[j]);
      }
    }

    // ---- gh = h @ Whh^T : A from LDS re-layout buffer -------------------
    #pragma unroll
    for (int k0 = 0; k0 < HQ; k0 += 4) {
      v2f ha = *(const v2f*)(hb + m * HQ + k0 + koff);
      #pragma unroll
      for (int tt = 0; tt < 8; tt++) {
        v2f wb = *(const v2f*)(sWhh + (tt * 16 + m) * HQ + k0 + koff);
        accRZ[tt] = wmma4(ha, wb, accRZ[tt]);
      }
      #pragma unroll
      for (int j = 0; j < 4; j++) {
        v2f wb = *(const v2f*)(sWhh + ((8 + j) * 16 + m) * HQ + k0 + koff);
        accHN[j] = wmma4(ha, wb, accHN[j]);
      }
    }

    // ---- gates + masked h update; refresh LDS h buffer ------------------
    #pragma unroll
    for (int j = 0; j < 4; j++) {
      #pragma unroll
      for (int v = 0; v < 8; v++) {
        float r  = sigmoidf_fast(accRZ[j][v]);
        float z  = sigmoidf_fast(accRZ[4 + j][v]);
        float nn = tanhf_fast(accIN[j][v] + r * accHN[j][v]);
        float hn = (1.0f - z) * nn + z * hreg[j][v];
        hreg[j][v] = (t < durD[v]) ? hn : hreg[j][v];
        hb[(v + 8 * hi) * HQ + j * 16 + m] = hreg[j][v];   // D-layout -> (m,j)
      }
    }
    // same-wave LDS is in-order: next step's A reads see these stores.
  }

  // ---- emit h_fwd ---------------------------------------------------------
  #pragma unroll
  for (int j = 0; j < 4; j++)
    #pragma unroll
    for (int v = 0; v < 8; v++)
      out[(size_t)(seq0 + v + 8 * hi) * 128 + j * 16 + m] = hreg[j][v];

  // ================= backward GRU: single cell, h0 = 0 =====================
  // gh = b_hh_b (constant), so this is one 16x192x80 GEMM on x_last.
  __syncthreads();
  for (int i = tid; i < G3 * DMELQ; i += nthreads) sWih[i] = W_ih_b[i];
  __syncthreads();

  float biasRZb[8], biasINb[4], hnb[4];
  #pragma unroll
  for (int tt = 0; tt < 8; tt++) {
    int n = tt * 16 + m;
    biasRZb[tt] = sBias[2 * G3 + n] + sBias[3 * G3 + n];
  }
  #pragma unroll
  for (int j = 0; j < 4; j++) {
    int n = 128 + j * 16 + m;
    biasINb[j] = sBias[2 * G3 + n];
    hnb[j]     = sBias[3 * G3 + n];
  }

  #pragma unroll
  for (int tt = 0; tt < 8; tt++)
    #pragma unroll
    for (int v = 0; v < 8; v++) accRZ[tt][v] = biasRZb[tt];
  #pragma unroll
  for (int j = 0; j < 4; j++)
    #pragma unroll
    for (int v = 0; v < 8; v++) accIN[j][v] = biasINb[j];

  int framel = start_m + dm - 1; framel = framel < (TQ - 1) ? framel : (TQ - 1);
  const float* xrowl = melsB + (size_t)framel * DMELQ;
  #pragma unroll
  for (int k0 = 0; k0 < DMELQ; k0 += 4) {
    v2f xa = *(const v2f*)(xrowl + k0 + koff);
    xa.x = fmaxf(0.0f, fmaf(a1, fmaxf(0.0f, fmaf(a0, xa.x, c0)), c1));
    xa.y = fmaxf(0.0f, fmaf(a1, fmaxf(0.0f, fmaf(a0, xa.y, c0)), c1));
    #pragma unroll
    for (int tt = 0; tt < 8; tt++) {
      v2f wb = *(const v2f*)(sWih + (tt * 16 + m) * DMELQ + k0 + koff);
      accRZ[tt] = wmma4(xa, wb, accRZ[tt]);
    }
    #pragma unroll
    for (int j = 0; j < 4; j++) {
      v2f wb = *(const v2f*)(sWih + ((8 + j) * 16 + m) * DMELQ + k0 + koff);
      accIN[j] = wmma4(xa, wb, accIN[j]);
    }
  }

  #pragma unroll
  for (int j = 0; j < 4; j++) {
    #pragma unroll
    for (int v = 0; v < 8; v++) {
      float r  = sigmoidf_fast(accRZ[j][v]);
      float z  = sigmoidf_fast(accRZ[4 + j][v]);
      float nn = tanhf_fast(accIN[j][v] + r * hnb[j]);
      float hb_out = (1.0f - z) * nn;                      // + z*0
      out[(size_t)(seq0 + v + 8 * hi) * 128 + 64 + j * 16 + m] = hb_out;
    }
  }
}

extern "C" void kernel_launch(void* const* d_in, const int* in_sizes, int n_in,
                              void* d_out, int out_size, void* d_ws, size_t ws_size,
                              hipStream_t stream) {
  (void)in_sizes; (void)n_in; (void)d_ws; (void)ws_size; (void)out_size;
  const float* mels   = (const float*)d_in[0];
  const int*   dur    = (const int*)  d_in[1];
  // d_in[2] = max_dur scalar (compile-time constant DMAXQ)
  const float* conv_w = (const float*)d_in[3];
  const float* conv_b = (const float*)d_in[4];
  const float* bn_g   = (const float*)d_in[5];
  const float* bn_b   = (const float*)d_in[6];
  const float* bn_m   = (const float*)d_in[7];
  const float* bn_v   = (const float*)d_in[8];
  const float* W_ih_f = (const float*)d_in[9];
  const float* W_hh_f = (const float*)d_in[10];
  const float* b_ih_f = (const float*)d_in[11];
  const float* b_hh_f = (const float*)d_in[12];
  const float* W_ih_b = (const float*)d_in[13];
  // d_in[14] = W_hh_b : unused (h0 == 0 makes h@W_hh_b == 0)
  const float* b_ih_b = (const float*)d_in[15];
  const float* b_hh_b = (const float*)d_in[16];
  float* out = (float*)d_out;

  const size_t shmem = (size_t)LDS_FLOATS * sizeof(float);   // 130,048 B
  hipFuncSetAttribute((const void*)prosody_gru_kernel,
                      hipFuncAttributeMaxDynamicSharedMemorySize, (int)shmem);

  const int tiles  = (BQ * PQ) / 16;        // 384
  const int blocks = tiles / WAVES;         // 96
  prosody_gru_kernel<<<dim3(blocks), dim3(WAVES * 32), shmem, stream>>>(
      mels, dur, conv_w, conv_b, bn_g, bn_b, bn_m, bn_v,
      W_ih_f, W_hh_f, b_ih_f, b_hh_f, W_ih_b, b_ih_b, b_hh_b, out);
}